// EmbeddingVoxel_41961830482624
// MI455X (gfx1250) — compile-verified
//
#include <hip/hip_runtime.h>
#include <math.h>

// EmbeddingVoxel forward: trilinear voxel-feature gather + positional encodings.
// out0: (N, 351) = [voxel_ftr(32) | sin/cos posenc x4 freqs (256) | xyz(3) | sin/cos posenc x10 freqs (60)]
// out1: (N, 8)   = voxel_ftr[:, 24:32]

#define TPB        256
#define CH         32
#define ACC_STRIDE 33      // LDS row stride (gcd(33,64)==1 -> bank-conflict-free)
#define OUT0_C     351
#define VF_C       288

typedef __attribute__((address_space(3))) float lds_f;

__device__ __forceinline__ float out0_elem(int p, int j,
                                           const float* __restrict__ sacc,
                                           const float* __restrict__ sxyz) {
    if (j < VF_C) {
        if (j < CH) return sacc[p * ACC_STRIDE + j];
        int t  = j - CH;
        int c  = t & (CH - 1);
        int fs = t >> 5;                       // fs = 2*freq + (0=sin, 1=cos)
        float x = sacc[p * ACC_STRIDE + c] * (float)(1 << (fs >> 1));
        // |x| <= ~8 rad: well inside v_sin_f32's valid range -> native trans ops
        return (fs & 1) ? __cosf(x) : __sinf(x);
    }
    int t = j - VF_C;
    if (t < 3) return sxyz[p * 3 + t];
    t -= 3;
    int fs = t / 3;
    int d  = t - fs * 3;
    float x = sxyz[p * 3 + d] * (float)(1 << (fs >> 1));
    // args up to ~1670 rad (265 revolutions) exceed native-sin range -> accurate libm
    return (fs & 1) ? cosf(x) : sinf(x);
}

__global__ __launch_bounds__(TPB)
void voxel_embed_fwd(const float* __restrict__ xyz,
                     const float* __restrict__ table,
                     const int*   __restrict__ vmap,
                     const float* __restrict__ voff,
                     const float* __restrict__ vsize,
                     const int*   __restrict__ vshape,
                     float* __restrict__ out0,
                     float* __restrict__ out1,
                     int N) {
    __shared__ float sacc[TPB * ACC_STRIDE];              // 33.0 KB
    __shared__ float sxyz[TPB * 3];                       //  3.0 KB
    __shared__ __align__(16) float s_inst[TPB * 8];       //  8.0 KB, layout == out1 tile

    const int tid     = threadIdx.x;
    const int base    = blockIdx.x * TPB;
    const int n       = base + tid;
    const int tilePts = min(TPB, N - base);

    // ---------------- Phase 1: gather + trilinear blend (1 thread / point) ----
    if (tid < tilePts) {
        const float ox = voff[0], oy = voff[1], oz = voff[2];
        const float vs = vsize[0];
        const int sx = vshape[0], sy = vshape[1], sz = vshape[2];

        const float px = xyz[(size_t)n * 3 + 0];
        const float py = xyz[(size_t)n * 3 + 1];
        const float pz = xyz[(size_t)n * 3 + 2];

        const float xsx = (px + ox) / vs;
        const float xsy = (py + oy) / vs;
        const float xsz = (pz + oz) / vs;

        const int q0x = (int)floorf(xsx);
        const int q0y = (int)floorf(xsy);
        const int q0z = (int)floorf(xsz);

        int   emb[8];
        float wts[8];
        #pragma unroll
        for (int k = 0; k < 8; ++k) {
            const int cx = q0x + ((k >> 2) & 1);
            const int cy = q0y + ((k >> 1) & 1);
            const int cz = q0z + (k & 1);
            const bool valid = (cx >= 0) & (cy >= 0) & (cz >= 0) &
                               (cx < sx) & (cy < sy) & (cz < sz);
            int e = -1;
            if (valid) e = vmap[((size_t)cx * sy + cy) * sz + cz];
            emb[k] = e;
            const float wx = 1.0f - fabsf(xsx - (float)cx);
            const float wy = 1.0f - fabsf(xsy - (float)cy);
            const float wz = 1.0f - fabsf(xsz - (float)cz);
            wts[k] = wx * wy * wz;
            // gfx1250 global_prefetch_b8: warm the 128B row while weight VALU runs,
            // breaking the dependent index->row latency chain.
            if (e >= 0) __builtin_prefetch(table + (size_t)e * CH, 0, 0);
        }

        float acc[CH];
        #pragma unroll
        for (int c = 0; c < CH; ++c) acc[c] = 0.0f;

        #pragma unroll
        for (int k = 0; k < 8; ++k) {
            if (emb[k] >= 0) {                 // invalid corners contribute 0
                const float4* row = (const float4*)(table + (size_t)emb[k] * CH);
                const float w = wts[k];
                #pragma unroll
                for (int t = 0; t < CH / 4; ++t) {   // 8x global_load_b128
                    const float4 v = row[t];
                    acc[4 * t + 0] += w * v.x;
                    acc[4 * t + 1] += w * v.y;
                    acc[4 * t + 2] += w * v.z;
                    acc[4 * t + 3] += w * v.w;
                }
            }
        }

        #pragma unroll
        for (int c = 0; c < CH; ++c) sacc[tid * ACC_STRIDE + c] = acc[c];
        #pragma unroll
        for (int j = 0; j < 8; ++j) s_inst[tid * 8 + j] = acc[24 + j];
        sxyz[tid * 3 + 0] = px;
        sxyz[tid * 3 + 1] = py;
        sxyz[tid * 3 + 2] = pz;
    }
    __syncthreads();

    // ---------------- Phase 2a: inst_ftr via async LDS->global (ASYNCcnt path) -
    // s_inst's flat layout equals the out1 tile layout, so each lane fires two
    // global_store_async_from_lds_b128: 16B straight from LDS to a coalesced
    // global address, no VGPR round-trip, overlapped with phase-2b VALU below.
    {
        float* o1 = out1 + (size_t)base * 8;
        if (tilePts == TPB && ((((uintptr_t)o1) & 15) == 0)) {
            float4* o4 = (float4*)o1;
            for (int i = tid; i < (TPB * 8) / 4; i += TPB) {   // 2 iterations
                const float4* gdst = o4 + i;
                lds_f* lsrc = (lds_f*)&s_inst[i * 4];          // 16B-aligned LDS src
                asm volatile("global_store_async_from_lds_b128 %0, %1, off"
                             :: "v"(gdst), "v"(lsrc)
                             : "memory");
            }
        } else {
            for (int i = tid; i < tilePts * 8; i += TPB) {
                o1[i] = s_inst[i];
            }
        }
    }

    // ---------------- Phase 2b: 351-wide row, block-contiguous float4 stores --
    {
        float* o0 = out0 + (size_t)base * OUT0_C;
        if (tilePts == TPB && ((((uintptr_t)o0) & 15) == 0)) {
            float4* o4 = (float4*)o0;
            const int tot4 = (TPB * OUT0_C) / 4;   // 22464 (256*351 % 4 == 0)
            for (int i = tid; i < tot4; i += TPB) {
                const int flat = i * 4;
                float4 v;
                float* vp = (float*)&v;
                #pragma unroll
                for (int u = 0; u < 4; ++u) {
                    const int f = flat + u;
                    const int p = f / OUT0_C;       // magic-multiply, cheap
                    const int j = f - p * OUT0_C;
                    vp[u] = out0_elem(p, j, sacc, sxyz);
                }
                o4[i] = v;
            }
        } else {
            for (int i = tid; i < tilePts * OUT0_C; i += TPB) {
                const int p = i / OUT0_C;
                const int j = i - p * OUT0_C;
                o0[i] = out0_elem(p, j, sacc, sxyz);
            }
        }
    }

    // Drain outstanding async LDS->global stores before LDS is deallocated.
    // (s_endpgm also implies wait-idle; this makes the ordering explicit.)
    asm volatile("s_wait_asynccnt 0x0" ::: "memory");
}

extern "C" void kernel_launch(void* const* d_in, const int* in_sizes, int n_in,
                              void* d_out, int out_size, void* d_ws, size_t ws_size,
                              hipStream_t stream) {
    const float* xyz    = (const float*)d_in[0];   // (N, 3) f32
    const float* table  = (const float*)d_in[1];   // (800000, 32) f32
    const int*   vmap   = (const int*)  d_in[2];   // (160,160,160) i32
    const float* voff   = (const float*)d_in[3];   // (3,) f32
    const float* vsize  = (const float*)d_in[4];   // (1,) f32
    const int*   vshape = (const int*)  d_in[5];   // (3,) i32

    const int N = in_sizes[0] / 3;
    float* out0 = (float*)d_out;                   // (N, 351)
    float* out1 = out0 + (size_t)N * OUT0_C;       // (N, 8)

    const int blocks = (N + TPB - 1) / TPB;
    voxel_embed_fwd<<<blocks, TPB, 0, stream>>>(xyz, table, vmap, voff, vsize,
                                                vshape, out0, out1, N);
}